// S6Layer_48679159333168
// MI455X (gfx1250) — compile-verified
//
#include <hip/hip_runtime.h>
#include <hip/hip_bf16.h>
#include <math.h>

// ---- problem constants (from reference) ----
#define BB      2
#define LSEQ    1024
#define DMODEL  768
#define DM      1536
#define DS      16
#define DTRANK  48
#define NXD     80        // DT_RANK + 2*D_STATE
#define NCHUNK  16
#define CHUNK   64        // LSEQ / NCHUNK
#define BL      (BB*LSEQ) // 2048 rows

typedef float v2f __attribute__((ext_vector_type(2)));
typedef float v8f __attribute__((ext_vector_type(8)));

// explicit global-address-space pointers so fragment loads lower to
// global_load_* (flat_load_* would also consume DScnt / the LDS pipe)
typedef const float __attribute__((address_space(1)))* gfp;
typedef const v2f   __attribute__((address_space(1)))* gv2fp;

__device__ __forceinline__ float sigmoidf_(float x) {
    return 1.0f / (1.0f + __expf(-x));
}
__device__ __forceinline__ float softplusf_(float x) {
    return (x > 20.0f) ? x : log1pf(__expf(x));
}

// ---------------------------------------------------------------------------
// WMMA fp32 GEMM: C[M,N] = A[M,K] @ B[K,N] (row-major, lda/ldb/ldc strides).
// Each wave computes a (MT*16) x (NT*16) tile with V_WMMA_F32_16X16X4_F32.
// Software double-buffered: fragments for step k+4 are loaded before the
// WMMAs of step k issue, so s_wait_loadcnt can retire loads lazily (overlap).
// Base pointers are fixed (addrspace(1)); only scalar offsets advance.
// Block = 256 threads = 8 waves stacked along M.
// Requires: M % (8*MT*16) == 0, N % (NT*16) == 0, K % 4 == 0, K >= 8.
// epilogue: 0 = none, 1 = softplus(v + bias[n])  (for the dt projection)
// ---------------------------------------------------------------------------
template <int MT, int NT>
__global__ void gemm_wmma(const float* __restrict__ A, const float* __restrict__ B,
                          float* __restrict__ C, const float* __restrict__ bias,
                          int K, int lda, int ldb, int ldc, int epilogue)
{
    const int wave = threadIdx.x >> 5;      // 0..7 (uniform per wave)
    const int lane = threadIdx.x & 31;
    const int half = lane >> 4;             // 0: K=k0,k0+1   1: K=k0+2,k0+3
    const int r    = lane & 15;
    const int m0   = (blockIdx.y * 8 + wave) * (MT * 16);
    const int n0   = blockIdx.x * (NT * 16);
    const int kh   = half * 2;

    v8f acc[MT][NT];
#pragma unroll
    for (int mt = 0; mt < MT; ++mt)
#pragma unroll
        for (int nt = 0; nt < NT; ++nt)
            acc[mt][nt] = (v8f){0,0,0,0,0,0,0,0};

    // A fragment: lane r holds A[m][k0+kh], A[m][k0+kh+1]   (16x4 layout)
    // B fragment: lane r holds B[k0+kh][n], B[k0+kh+1][n]   (4x16 layout)
    gfp a0[MT];
#pragma unroll
    for (int mt = 0; mt < MT; ++mt)
        a0[mt] = (gfp)(A + (size_t)(m0 + mt * 16 + r) * lda + kh);
    gfp b0 = (gfp)(B + (size_t)kh * ldb + n0 + r);

    v2f a_cur[MT], b_cur[NT];
#pragma unroll
    for (int mt = 0; mt < MT; ++mt) a_cur[mt] = *(gv2fp)a0[mt];
#pragma unroll
    for (int nt = 0; nt < NT; ++nt) {
        b_cur[nt].x = b0[nt * 16];
        b_cur[nt].y = b0[nt * 16 + ldb];
    }

    int ka = 0;   // element offset along A rows
    int kb = 0;   // element offset along B (multiples of ldb)
    for (int k0 = 0; k0 < K - 4; k0 += 4) {
        // prefetch k0+4 fragments (independent of the WMMAs below)
        v2f a_nxt[MT], b_nxt[NT];
#pragma unroll
        for (int mt = 0; mt < MT; ++mt)
            a_nxt[mt] = *(gv2fp)(a0[mt] + ka + 4);
        const int kb4 = kb + 4 * ldb;
#pragma unroll
        for (int nt = 0; nt < NT; ++nt) {
            b_nxt[nt].x = b0[kb4 + nt * 16];
            b_nxt[nt].y = b0[kb4 + nt * 16 + ldb];
        }
        // compute on current fragments
#pragma unroll
        for (int mt = 0; mt < MT; ++mt)
#pragma unroll
            for (int nt = 0; nt < NT; ++nt)
                acc[mt][nt] = __builtin_amdgcn_wmma_f32_16x16x4_f32(
                    false, a_cur[mt], false, b_cur[nt], (short)0, acc[mt][nt],
                    false, false);
        // rotate
#pragma unroll
        for (int mt = 0; mt < MT; ++mt) a_cur[mt] = a_nxt[mt];
#pragma unroll
        for (int nt = 0; nt < NT; ++nt) b_cur[nt] = b_nxt[nt];
        ka += 4;
        kb = kb4;
    }
    // peeled last k-step
#pragma unroll
    for (int mt = 0; mt < MT; ++mt)
#pragma unroll
        for (int nt = 0; nt < NT; ++nt)
            acc[mt][nt] = __builtin_amdgcn_wmma_f32_16x16x4_f32(
                false, a_cur[mt], false, b_cur[nt], (short)0, acc[mt][nt],
                false, false);

    // C/D 16x16 f32 layout: VGPR i -> row +8*half + i, col n0 + 16*nt + r
#pragma unroll
    for (int mt = 0; mt < MT; ++mt) {
#pragma unroll
        for (int nt = 0; nt < NT; ++nt) {
            const int nc = n0 + nt * 16 + r;
            float bv = (epilogue == 1) ? bias[nc] : 0.0f;
#pragma unroll
            for (int i = 0; i < 8; ++i) {
                float v = acc[mt][nt][i];
                if (epilogue == 1) v = softplusf_(v + bv);
                C[(size_t)(m0 + mt * 16 + half * 8 + i) * ldc + nc] = v;
            }
        }
    }
}

// ---------------------------------------------------------------------------
// Depthwise conv1d (K=4, pad (1,2)) + SiLU on the x_m half of xz.
// ---------------------------------------------------------------------------
__global__ void conv_silu_kernel(const float* __restrict__ xz,
                                 const float* __restrict__ w,   // (4,1,1536)
                                 const float* __restrict__ bias,
                                 float* __restrict__ xc)
{
    int idx = blockIdx.x * blockDim.x + threadIdx.x;   // over BL*DM
    if (idx >= BL * DM) return;
    int c  = idx % DM;
    int bl = idx / DM;
    int l  = bl % LSEQ;
    int b  = bl / LSEQ;

    float acc = bias[c];
#pragma unroll
    for (int k = 0; k < 4; ++k) {
        int ls = l - 1 + k;
        if (ls >= 0 && ls < LSEQ)
            acc += w[k * DM + c] * xz[((size_t)(b * LSEQ + ls)) * (2 * DM) + c];
    }
    xc[idx] = acc * sigmoidf_(acc);   // silu
}

// ---------------------------------------------------------------------------
// Selective scan, pass A: per-chunk local scan with zero initial state.
// Block = 256 channels of one (b,chunk); 6 blocks cover DM=1536.
// Writes y_part = sum_s h_local*C, and chunk carries (h_end, Pfull).
// ---------------------------------------------------------------------------
__global__ void scan_passA(const float* __restrict__ dt,
                           const float* __restrict__ xc,
                           const float* __restrict__ xdbl,   // (BL,80): [48:64]=B, [64:80]=C
                           const float* __restrict__ A_log,
                           float* __restrict__ y_part,
                           float* __restrict__ carryH,
                           float* __restrict__ carryP)
{
    const int nblk  = blockIdx.x % (DM / 256);
    const int chunk = (blockIdx.x / (DM / 256)) % NCHUNK;
    const int b     = blockIdx.x / ((DM / 256) * NCHUNK);
    const int n     = nblk * 256 + threadIdx.x;

    __shared__ float sB[DS];
    __shared__ float sC[DS];

    float negA[DS], h[DS], P[DS];
#pragma unroll
    for (int s = 0; s < DS; ++s) {
        negA[s] = -__expf(A_log[n * DS + s]);
        h[s] = 0.0f;
        P[s] = 1.0f;
    }

    const int l0 = chunk * CHUNK;
    for (int li = 0; li < CHUNK; ++li) {
        const int bl = b * LSEQ + l0 + li;
        __syncthreads();
        if (threadIdx.x < 32) {
            int s = threadIdx.x & 15;
            if (threadIdx.x < 16) sB[s] = xdbl[(size_t)bl * NXD + DTRANK + s];
            else                  sC[s] = xdbl[(size_t)bl * NXD + DTRANK + DS + s];
        }
        __syncthreads();

        const float dtv = dt[(size_t)bl * DM + n];
        const float xv  = xc[(size_t)bl * DM + n];
        const float dx  = dtv * xv;
        float y = 0.0f;
#pragma unroll
        for (int s = 0; s < DS; ++s) {
            float da = __expf(dtv * negA[s]);
            h[s] = da * h[s] + dx * sB[s];
            P[s] *= da;
            y += h[s] * sC[s];
        }
        y_part[(size_t)bl * DM + n] = y;
    }

    const size_t ci = (((size_t)b * NCHUNK + chunk) * DM + n) * DS;
#pragma unroll
    for (int s = 0; s < DS; ++s) {
        carryH[ci + s] = h[s];
        carryP[ci + s] = P[s];
    }
}

// ---------------------------------------------------------------------------
// Chain the 16 chunk carries: cin[chunk] = H_{chunk-1},
// H_c = Pfull_c * H_{c-1} + h_end_c.  One thread per (b,n,s) = 49152 threads.
// ---------------------------------------------------------------------------
__global__ void carry_combine(const float* __restrict__ carryH,
                              const float* __restrict__ carryP,
                              float* __restrict__ cin)
{
    int t = blockIdx.x * blockDim.x + threadIdx.x;
    if (t >= BB * DM * DS) return;
    int s = t % DS;
    int n = (t / DS) % DM;
    int b = t / (DS * DM);
    float H = 0.0f;
    for (int c = 0; c < NCHUNK; ++c) {
        size_t idx = (((size_t)b * NCHUNK + c) * DM + n) * DS + s;
        cin[idx] = H;
        H = carryP[idx] * H + carryH[idx];
    }
}

// ---------------------------------------------------------------------------
// Selective scan, pass B: apply carry correction + x*D + silu(z) gate.
// y[l] = (y_part[l] + sum_s Pcum[l,s]*cin[s]*C[l,s] + x*D[n]) * silu(z)
// where Pcum[l] = prod_{j<=l within chunk} dA[j] (dA recomputed).
// ---------------------------------------------------------------------------
__global__ void scan_passB(const float* __restrict__ dt,
                           const float* __restrict__ xc,
                           const float* __restrict__ xz,     // z = xz[..., 1536:]
                           const float* __restrict__ xdbl,
                           const float* __restrict__ A_log,
                           const float* __restrict__ Dvec,
                           const float* __restrict__ cin,
                           float* __restrict__ y_g)
{
    const int nblk  = blockIdx.x % (DM / 256);
    const int chunk = (blockIdx.x / (DM / 256)) % NCHUNK;
    const int b     = blockIdx.x / ((DM / 256) * NCHUNK);
    const int n     = nblk * 256 + threadIdx.x;

    __shared__ float sC[DS];

    float negA[DS], P[DS], ci[DS];
    const size_t cidx = (((size_t)b * NCHUNK + chunk) * DM + n) * DS;
#pragma unroll
    for (int s = 0; s < DS; ++s) {
        negA[s] = -__expf(A_log[n * DS + s]);
        P[s] = 1.0f;
        ci[s] = cin[cidx + s];
    }
    const float Dn = Dvec[n];

    const int l0 = chunk * CHUNK;
    for (int li = 0; li < CHUNK; ++li) {
        const int bl = b * LSEQ + l0 + li;
        __syncthreads();
        if (threadIdx.x < 16)
            sC[threadIdx.x] = xdbl[(size_t)bl * NXD + DTRANK + DS + threadIdx.x];
        __syncthreads();

        const float dtv = dt[(size_t)bl * DM + n];
        const float xv  = xc[(size_t)bl * DM + n];
        const float zv  = xz[(size_t)bl * (2 * DM) + DM + n];

        float corr = 0.0f;
#pragma unroll
        for (int s = 0; s < DS; ++s) {
            float da = __expf(dtv * negA[s]);
            P[s] *= da;
            corr += P[s] * ci[s] * sC[s];
        }
        float y = y_g[(size_t)bl * DM + n] + corr + xv * Dn;
        y_g[(size_t)bl * DM + n] = y * (zv * sigmoidf_(zv));
    }
}

// ---------------------------------------------------------------------------
// launcher
// ---------------------------------------------------------------------------
extern "C" void kernel_launch(void* const* d_in, const int* in_sizes, int n_in,
                              void* d_out, int out_size, void* d_ws, size_t ws_size,
                              hipStream_t stream) {
    const float* x      = (const float*)d_in[0];  // (B,L,768)
    const float* W_in   = (const float*)d_in[1];  // (768,3072)
    const float* conv_w = (const float*)d_in[2];  // (4,1,1536)
    const float* conv_b = (const float*)d_in[3];  // (1536)
    const float* A_log  = (const float*)d_in[4];  // (1536,16)
    const float* Dvec   = (const float*)d_in[5];  // (1536)
    const float* W_x    = (const float*)d_in[6];  // (1536,80)
    const float* W_dt   = (const float*)d_in[7];  // (48,1536)
    const float* b_dt   = (const float*)d_in[8];  // (1536)
    const float* W_out  = (const float*)d_in[9];  // (1536,768)
    float* out = (float*)d_out;                   // (B,L,768)

    // workspace layout (floats)
    float* ws   = (float*)d_ws;
    float* xz   = ws;                                   // BL*3072
    float* xc   = xz   + (size_t)BL * 2 * DM;           // BL*1536
    float* xdbl = xc   + (size_t)BL * DM;               // BL*80
    float* dtb  = xdbl + (size_t)BL * NXD;              // BL*1536
    float* y_g  = dtb  + (size_t)BL * DM;               // BL*1536
    float* cH   = y_g  + (size_t)BL * DM;               // B*NCHUNK*DM*DS
    float* cP   = cH   + (size_t)BB * NCHUNK * DM * DS;
    float* cin  = cP   + (size_t)BB * NCHUNK * DM * DS;

    const dim3 blk(256);

    // 1) xz = x @ W_in          M=2048 K=768  N=3072   (32x64 wave tiles)
    gemm_wmma<2, 4><<<dim3((2 * DM) / 64, BL / 256), blk, 0, stream>>>(
        x, W_in, xz, nullptr, DMODEL, DMODEL, 2 * DM, 2 * DM, 0);

    // 2) x_c = silu(conv1d(x_m))
    conv_silu_kernel<<<dim3((BL * DM + 255) / 256), blk, 0, stream>>>(
        xz, conv_w, conv_b, xc);

    // 3) x_dbl = x_c @ W_x      M=2048 K=1536 N=80     (32x16 wave tiles)
    gemm_wmma<2, 1><<<dim3(NXD / 16, BL / 256), blk, 0, stream>>>(
        xc, W_x, xdbl, nullptr, DM, DM, NXD, NXD, 0);

    // 4) dt = softplus(x_dbl[:, :48] @ W_dt + b_dt)   M=2048 K=48 N=1536
    gemm_wmma<2, 4><<<dim3(DM / 64, BL / 256), blk, 0, stream>>>(
        xdbl, W_dt, dtb, b_dt, DTRANK, NXD, DM, DM, 1);

    // 5) chunked selective scan
    const int nScanBlocks = BB * NCHUNK * (DM / 256);   // 192
    scan_passA<<<dim3(nScanBlocks), blk, 0, stream>>>(
        dtb, xc, xdbl, A_log, y_g, cH, cP);
    carry_combine<<<dim3((BB * DM * DS + 255) / 256), blk, 0, stream>>>(cH, cP, cin);
    scan_passB<<<dim3(nScanBlocks), blk, 0, stream>>>(
        dtb, xc, xz, xdbl, A_log, Dvec, cin, y_g);

    // 6) out = y_g @ W_out      M=2048 K=1536 N=768
    gemm_wmma<2, 4><<<dim3(DMODEL / 64, BL / 256), blk, 0, stream>>>(
        y_g, W_out, out, nullptr, DM, DM, DMODEL, DMODEL, 0);
}